// BAGDnet_85229331021876
// MI455X (gfx1250) — compile-verified
//
#include <hip/hip_runtime.h>
#include <stdint.h>

// ---------------------------------------------------------------------------
// CDNA5 async global->LDS copy helpers (ASYNCcnt-tracked, gfx1250-specific).
// Builtin signature (probe-confirmed via diagnostic): typed v4i pointers,
// (global src, lds dst, imm offset, imm cpol). Inline-asm fallback per
// CDNA5 ISA if the builtin is absent.
// ---------------------------------------------------------------------------
#define GLOBAL_AS __attribute__((address_space(1)))
#define LOCAL_AS  __attribute__((address_space(3)))

typedef int v4i_gcc __attribute__((vector_size(16)));

static __device__ __forceinline__ void async_copy_b128(const void* gsrc, void* lds_dst)
{
#if __has_builtin(__builtin_amdgcn_global_load_async_to_lds_b128)
    __builtin_amdgcn_global_load_async_to_lds_b128(
        (GLOBAL_AS v4i_gcc*)(uintptr_t)gsrc,
        (LOCAL_AS  v4i_gcc*)(uint32_t)(uintptr_t)lds_dst,  // low 32 bits of flat LDS addr = LDS offset
        0, 0);
#else
    uint32_t lds_off = (uint32_t)(uintptr_t)lds_dst;
    uint64_t ga      = (uint64_t)(uintptr_t)gsrc;
    asm volatile("global_load_async_to_lds_b128 %0, %1, off"
                 :: "v"(lds_off), "v"(ga) : "memory");
#endif
}

static __device__ __forceinline__ void wait_async_zero()
{
#if __has_builtin(__builtin_amdgcn_s_wait_asynccnt)
    __builtin_amdgcn_s_wait_asynccnt(0);
#else
    asm volatile("s_wait_asynccnt 0" ::: "memory");
#endif
}

// ---------------------------------------------------------------------------
// Kernel 0: zero-fill inverse permutation tables (reference initializes to 0).
// ---------------------------------------------------------------------------
__global__ void k_zero(int* __restrict__ inv_kf, int* __restrict__ inv_mp,
                       int nkf, int nmp)
{
    int i = blockIdx.x * blockDim.x + threadIdx.x;
    if (i < nkf) inv_kf[i] = 0;
    if (i < nmp) inv_mp[i] = 0;
}

// ---------------------------------------------------------------------------
// Kernel 1: scatter inverse permutations: inv[idx[i]] = i
// ---------------------------------------------------------------------------
__global__ void k_scatter(const int* __restrict__ idxKF, const int* __restrict__ idxMP,
                          int* __restrict__ inv_kf, int* __restrict__ inv_mp,
                          int nkf, int nmp)
{
    int i = blockIdx.x * blockDim.x + threadIdx.x;
    if (i < nkf) inv_kf[idxKF[i]] = i;
    if (i < nmp) inv_mp[idxMP[i]] = i;
}

// ---------------------------------------------------------------------------
// Kernel 2: quat-log -> rotation, pack pose as 4 x float4 per frame:
//   row r = (R[r][0], R[r][1], R[r][2], KFXYZ[r]),  row3 = (0,0,0,KFXYZ[3])
// 64 B per frame -> nkf*64 B table in workspace.
// ---------------------------------------------------------------------------
__global__ void k_pose(const float* __restrict__ qlogs,
                       const float* __restrict__ kfxyz,
                       float4* __restrict__ poseT, int nkf)
{
    int i = blockIdx.x * blockDim.x + threadIdx.x;
    if (i >= nkf) return;

    float qx = qlogs[3 * i + 0];
    float qy = qlogs[3 * i + 1];
    float qz = qlogs[3 * i + 2];

    float n  = sqrtf(qx * qx + qy * qy + qz * qz);
    float ns = fmaxf(n, 1e-8f);
    float w  = cosf(n);
    float s  = sinf(n) / ns;
    float x = qx * s, y = qy * s, z = qz * s;

    // reference re-normalizes the quaternion
    float qn = sqrtf(w * w + x * x + y * y + z * z);
    float r  = 1.0f / fmaxf(qn, 1e-8f);
    w *= r; x *= r; y *= r; z *= r;

    float tx = kfxyz[4 * i + 0];
    float ty = kfxyz[4 * i + 1];
    float tz = kfxyz[4 * i + 2];
    float tw = kfxyz[4 * i + 3];

    poseT[4 * i + 0] = make_float4(1.0f - 2.0f * (y * y + z * z),
                                   2.0f * (x * y - w * z),
                                   2.0f * (x * z + w * y), tx);
    poseT[4 * i + 1] = make_float4(2.0f * (x * y + w * z),
                                   1.0f - 2.0f * (x * x + z * z),
                                   2.0f * (y * z - w * x), ty);
    poseT[4 * i + 2] = make_float4(2.0f * (x * z - w * y),
                                   2.0f * (y * z + w * x),
                                   1.0f - 2.0f * (x * x + y * y), tz);
    poseT[4 * i + 3] = make_float4(0.0f, 0.0f, 0.0f, tw);
}

// ---------------------------------------------------------------------------
// Kernel 3: main projection. Persistent grid; each block async-stages the
// full pose table (nkf * 64 B) into LDS, then grid-strides over observations.
// ---------------------------------------------------------------------------
__global__ __launch_bounds__(256)
void k_project(const float4* __restrict__ poseT,
               const int* __restrict__ inv_kf,
               const int* __restrict__ inv_mp,
               const float4* __restrict__ pts,
               const float* __restrict__ Kmat,
               const int* __restrict__ frame_id,
               const int* __restrict__ point_id,
               float2* __restrict__ out,
               int nkf, int M)
{
    extern __shared__ float4 ldsT[];   // nkf * 4 float4

    // --- async global -> LDS staging of the pose table (ASYNCcnt path) ---
    int nvec = nkf * 4;
    for (int i = (int)threadIdx.x; i < nvec; i += (int)blockDim.x)
        async_copy_b128(&poseT[i], &ldsT[i]);
    wait_async_zero();
    __syncthreads();

    // camera intrinsics (uniform -> scalar loads)
    float k00 = Kmat[0], k01 = Kmat[1], k02 = Kmat[2];
    float k10 = Kmat[3], k11 = Kmat[4], k12 = Kmat[5];
    float k20 = Kmat[6], k21 = Kmat[7], k22 = Kmat[8];

    int stride = (int)(gridDim.x * blockDim.x);
    for (int m = (int)(blockIdx.x * blockDim.x + threadIdx.x); m < M; m += stride) {
        int f = inv_kf[frame_id[m]];
        int p = inv_mp[point_id[m]];

        float4 pt = pts[p];                 // global_load_b128 gather (L2-resident)
        float4 r0 = ldsT[f * 4 + 0];        // ds_load_b128
        float4 r1 = ldsT[f * 4 + 1];
        float4 r2 = ldsT[f * 4 + 2];
        float  w3 = ldsT[f * 4 + 3].w;

        float p0 = r0.x * pt.x + r0.y * pt.y + r0.z * pt.z + r0.w * pt.w;
        float p1 = r1.x * pt.x + r1.y * pt.y + r1.z * pt.z + r1.w * pt.w;
        float p2 = r2.x * pt.x + r2.y * pt.y + r2.z * pt.z + r2.w * pt.w;
        float p3 = w3 * pt.w;

        float c0 = p0 / p3, c1 = p1 / p3, c2 = p2 / p3;

        float u = c0 * k00 + c1 * k01 + c2 * k02;
        float v = c0 * k10 + c1 * k11 + c2 * k12;
        float z = c0 * k20 + c1 * k21 + c2 * k22;

        out[m] = make_float2(u / z, v / z);
    }
}

// ---------------------------------------------------------------------------
// Launcher
// ---------------------------------------------------------------------------
extern "C" void kernel_launch(void* const* d_in, const int* in_sizes, int n_in,
                              void* d_out, int out_size, void* d_ws, size_t ws_size,
                              hipStream_t stream)
{
    const float* KFquatlogs = (const float*)d_in[0];
    const float* KFXYZ      = (const float*)d_in[1];
    const float* tMPhomo    = (const float*)d_in[2];
    const float* Kmat       = (const float*)d_in[3];
    const int*   frame_id   = (const int*)d_in[4];
    const int*   point_id   = (const int*)d_in[5];
    const int*   idxKF      = (const int*)d_in[6];
    const int*   idxMP      = (const int*)d_in[7];

    int nkf = in_sizes[0] / 3;
    int nmp = in_sizes[2] / 4;
    int M   = in_sizes[4];

    // workspace layout: [poseT : nkf*64 B][inv_kf : nkf int][inv_mp : nmp int]
    char*   ws     = (char*)d_ws;
    float4* poseT  = (float4*)ws;
    int*    inv_kf = (int*)(ws + (size_t)nkf * 64);
    int*    inv_mp = inv_kf + nkf;

    const int T = 256;
    int nmax = (nkf > nmp) ? nkf : nmp;

    k_zero   <<<(nmax + T - 1) / T, T, 0, stream>>>(inv_kf, inv_mp, nkf, nmp);
    k_scatter<<<(nmax + T - 1) / T, T, 0, stream>>>(idxKF, idxMP, inv_kf, inv_mp, nkf, nmp);
    k_pose   <<<(nkf  + T - 1) / T, T, 0, stream>>>(KFquatlogs, KFXYZ, poseT, nkf);

    size_t lds_bytes = (size_t)nkf * 64;   // 128 KB for nkf=2000 -> 2 WGs/WGP (320 KB budget)
    int blocks = 512;                      // persistent-ish grid; grid-stride covers M
    k_project<<<blocks, T, lds_bytes, stream>>>(poseT, inv_kf, inv_mp,
                                                (const float4*)tMPhomo, Kmat,
                                                frame_id, point_id,
                                                (float2*)d_out, nkf, M);
}